// T1_37271726195189
// MI455X (gfx1250) — compile-verified
//
#include <hip/hip_runtime.h>

typedef float v2f __attribute__((ext_vector_type(2)));
typedef float v8f __attribute__((ext_vector_type(8)));

#define N_NODES 50000
#define PP 127           // previous_embed_size
#define AA 128           // agg features
#define OO 255           // output features
#define BN_EPS 1e-5f
#define MTILES 3125      // 50000 / 16

// ---------------- zero agg + stats ----------------
__global__ void zero_kernel(float* __restrict__ agg, float* __restrict__ stats) {
    size_t i = (size_t)blockIdx.x * blockDim.x + threadIdx.x;
    size_t total = (size_t)N_NODES * AA;
    size_t stride = (size_t)gridDim.x * blockDim.x;
    for (size_t p = i; p < total; p += stride) agg[p] = 0.0f;
    if (i < 2 * AA) stats[i] = 0.0f;
}

// ---------------- scatter-add of per-edge messages ----------------
// src_u(e) = cat(remember_u[e], g[e]) -> agg[v[e]] ; src_v(e) -> agg[u[e]]
__global__ void scatter_kernel(const int* __restrict__ u, const int* __restrict__ v,
                               const float* __restrict__ g,
                               const float* __restrict__ ru, const float* __restrict__ rv,
                               float* __restrict__ agg, int nev) {
    int e = blockIdx.x * 2 + (threadIdx.x >> 7);   // 2 events per 256-thread block
    int j = threadIdx.x & 127;
    if (e >= nev) return;
    float gg = g[e];
    float su = (j < PP) ? ru[(size_t)e * PP + j] : gg;
    float sv = (j < PP) ? rv[(size_t)e * PP + j] : gg;
    unsafeAtomicAdd(&agg[(size_t)v[e] * AA + j], su);   // global_atomic_add_f32
    unsafeAtomicAdd(&agg[(size_t)u[e] * AA + j], sv);
}

// ---------------- BN statistics: per-column sum / sumsq ----------------
__global__ void bnstats_kernel(const float* __restrict__ agg, float* __restrict__ stats) {
    int c = threadIdx.x;                 // 128 threads: one column each (coalesced rows)
    float s = 0.0f, s2 = 0.0f;
    for (int r = blockIdx.x; r < N_NODES; r += gridDim.x) {
        float val = agg[(size_t)r * AA + c];
        s += val; s2 += val * val;
    }
    unsafeAtomicAdd(&stats[c], s);
    unsafeAtomicAdd(&stats[AA + c], s2);
}

__global__ void bnfinal_kernel(const float* __restrict__ stats,
                               const float* __restrict__ gamma,
                               const float* __restrict__ beta,
                               float* __restrict__ ss) {
    int c = threadIdx.x;                 // 128 threads
    float inv_n = 1.0f / (float)N_NODES;
    float mean = stats[c] * inv_n;
    float var  = stats[AA + c] * inv_n - mean * mean;   // biased var (ddof=0)
    float scl  = gamma[c] * rsqrtf(var + BN_EPS);
    ss[c]      = scl;
    ss[AA + c] = beta[c] - mean * scl;
}

// ---------------- GEMM1: x1 = relu(BN(agg) @ w1_w^T + w1_b) ----------------
// fp32 WMMA 16x16x4. One wave per 16x16 tile; block = 8 waves covers all 8 N-tiles.
__global__ void gemm1_kernel(const float* __restrict__ agg,
                             const float* __restrict__ ss,     // scale[128], shift[128]
                             const float* __restrict__ w1w,    // [128][128] row-major
                             const float* __restrict__ w1b,
                             float* __restrict__ x1) {
    int mtile = blockIdx.x;              // 0..3124
    int ntile = threadIdx.x >> 5;        // wave id 0..7
    int lane  = threadIdx.x & 31;
    int grp   = lane >> 4;               // K-pair group (0: K+0/1, 1: K+2/3)
    int lr    = lane & 15;
    int arow  = mtile * 16 + lr;         // A fragment row (M)
    int bcol  = ntile * 16 + lr;         // B fragment col (N) ; B[k][n] = w1w[n][k]
    const float* scl = ss;
    const float* shf = ss + AA;

    v8f acc = {};
    for (int k0 = 0; k0 < AA; k0 += 4) {
        int k = k0 + grp * 2;
        v2f a, b;
        // BN applied as affine on the fly
        a.x = agg[(size_t)arow * AA + k]     * scl[k]     + shf[k];
        a.y = agg[(size_t)arow * AA + k + 1] * scl[k + 1] + shf[k + 1];
        b.x = w1w[(size_t)bcol * AA + k];
        b.y = w1w[(size_t)bcol * AA + k + 1];
        acc = __builtin_amdgcn_wmma_f32_16x16x4_f32(false, a, false, b,
                                                    (short)0, acc, false, false);
    }
    // D layout: VGPR r -> M = r + 8*grp, N = lr  (within tile)
    int n  = ntile * 16 + lr;
    int rb = mtile * 16 + grp * 8;
    float bias = w1b[n];
    #pragma unroll
    for (int r = 0; r < 8; ++r) {
        float val = acc[r] + bias;
        x1[(size_t)(rb + r) * AA + n] = fmaxf(val, 0.0f);
    }
}

// ---------------- GEMM2: out = cat(h, x1) @ w2_w^T + w2_b ----------------
// K and N padded 255 -> 256 with zeros; store masks column 255.
__device__ __forceinline__ float loadA2(const float* __restrict__ h,
                                        const float* __restrict__ x1,
                                        int row, int k) {
    if (k < PP) return h[(size_t)row * PP + k];
    if (k < OO) return x1[(size_t)row * AA + (k - PP)];
    return 0.0f;
}
__device__ __forceinline__ float loadB2(const float* __restrict__ w, int col, int k) {
    return (col < OO && k < OO) ? w[(size_t)col * OO + k] : 0.0f;
}

__global__ void gemm2_kernel(const float* __restrict__ h,
                             const float* __restrict__ x1,
                             const float* __restrict__ w2w,    // [255][255] row-major
                             const float* __restrict__ w2b,
                             float* __restrict__ out) {
    int mtile = blockIdx.x;                       // 0..3124
    int ntile = blockIdx.y * 8 + (threadIdx.x >> 5);  // 0..15
    int lane  = threadIdx.x & 31;
    int grp   = lane >> 4;
    int lr    = lane & 15;
    int arow  = mtile * 16 + lr;
    int bcol  = ntile * 16 + lr;

    v8f acc = {};
    for (int k0 = 0; k0 < 256; k0 += 4) {
        int k = k0 + grp * 2;
        v2f a, b;
        a.x = loadA2(h, x1, arow, k);
        a.y = loadA2(h, x1, arow, k + 1);
        b.x = loadB2(w2w, bcol, k);
        b.y = loadB2(w2w, bcol, k + 1);
        acc = __builtin_amdgcn_wmma_f32_16x16x4_f32(false, a, false, b,
                                                    (short)0, acc, false, false);
    }
    int n = ntile * 16 + lr;
    if (n < OO) {
        int rb = mtile * 16 + grp * 8;
        float bias = w2b[n];
        #pragma unroll
        for (int r = 0; r < 8; ++r)
            out[(size_t)(rb + r) * OO + n] = acc[r] + bias;
    }
}

extern "C" void kernel_launch(void* const* d_in, const int* in_sizes, int n_in,
                              void* d_out, int out_size, void* d_ws, size_t ws_size,
                              hipStream_t stream) {
    const int*   u  = (const int*)  d_in[0];
    const int*   v  = (const int*)  d_in[1];
    const float* g  = (const float*)d_in[2];
    const float* h  = (const float*)d_in[3];
    const float* ru = (const float*)d_in[4];
    const float* rv = (const float*)d_in[5];
    const float* gamma = (const float*)d_in[6];
    const float* beta  = (const float*)d_in[7];
    const float* w1w = (const float*)d_in[8];
    const float* w1b = (const float*)d_in[9];
    const float* w2w = (const float*)d_in[10];
    const float* w2b = (const float*)d_in[11];
    const int nev = in_sizes[0];

    // workspace layout
    float* agg   = (float*)d_ws;                       // [N_NODES][128]
    float* x1    = agg + (size_t)N_NODES * AA;         // [N_NODES][128]
    float* stats = x1  + (size_t)N_NODES * AA;         // sum[128], sumsq[128]
    float* ss    = stats + 2 * AA;                     // scale[128], shift[128]
    float* out   = (float*)d_out;                      // [N_NODES][255]

    zero_kernel<<<2048, 256, 0, stream>>>(agg, stats);
    scatter_kernel<<<(nev + 1) / 2, 256, 0, stream>>>(u, v, g, ru, rv, agg, nev);
    bnstats_kernel<<<512, 128, 0, stream>>>(agg, stats);
    bnfinal_kernel<<<1, 128, 0, stream>>>(stats, gamma, beta, ss);
    gemm1_kernel<<<MTILES, 256, 0, stream>>>(agg, ss, w1w, w1b, x1);
    gemm2_kernel<<<dim3(MTILES, 2), 256, 0, stream>>>(h, x1, w2w, w2b, out);
}